// LayerBlender_59889023975481
// MI455X (gfx1250) — compile-verified
//
#include <hip/hip_runtime.h>

typedef float v8f __attribute__((ext_vector_type(8)));
typedef float v2f __attribute__((ext_vector_type(2)));

#define LT 36        // teacher layers (conv in-channels)
#define LS 28        // student layers (conv out-channels)
#define KSZ 3        // conv kernel size
#define NPOS 16384   // B*H*S
#define DDIM 128     // head dim = conv length
#define NPAIR 18     // LT/2 layer pairs in LDS
#define PSTRIDE 288  // dwords per layer-pair row: 130 padded cols * 2, padded so
                     // (+PSTRIDE) == +32 banks -> lane halves on disjoint banks

#if __has_builtin(__builtin_amdgcn_global_load_async_to_lds_b32)
#define ASYNC_FILL 1
#else
#define ASYNC_FILL 0
#endif

__device__ __forceinline__ void wait_async_zero() {
#if __has_builtin(__builtin_amdgcn_s_wait_asynccnt)
    __builtin_amdgcn_s_wait_asynccnt(0);
#else
    asm volatile("s_wait_asynccnt 0" ::: "memory");
#endif
}

// One block = one (tensor, position-n) pair.
// LDS holds X pair-interleaved: Xp[kp][dpad] = {X[2kp][d], X[2kp+1][d]},
// zero halo at dpad=0 and dpad=129. Each B fragment is then ONE ds_load_b64.
// 4 waves do the 28x(36*3) layer mix as v_wmma_f32_16x16x4_f32:
// wave -> one M-tile (16 rows) x 4 d-tiles (16 cols each), 27 WMMAs per tile.
__global__ __launch_bounds__(128) void blend_wmma_f32(
    const float* __restrict__ ks, const float* __restrict__ vs,
    const float* __restrict__ wk, const float* __restrict__ wv,
    float* __restrict__ out)
{
    __shared__ float Xl[NPAIR * PSTRIDE];

    const int bid    = blockIdx.x;
    const int tensor = bid >> 14;          // 0 = K path, 1 = V path
    const int n      = bid & (NPOS - 1);

    const float* __restrict__ x = tensor ? vs : ks;
    const float* __restrict__ w = tensor ? wv : wk;
    float* __restrict__ o = out + (size_t)tensor * (size_t)LS * NPOS * DDIM;

    const int tid = threadIdx.x;

    // ---- fill: X[li][d] -> pair-interleaved padded LDS ----
    const float* xrow = x + (size_t)n * DDIM;
#if ASYNC_FILL
    // CDNA5 async DMA: per-lane global src, per-lane LDS dst -> the interleave
    // scatter happens inside the async unit, no VGPR round trip (ASYNCcnt).
    typedef __attribute__((address_space(1))) int GI;
    typedef __attribute__((address_space(3))) int LI;
    for (int i = tid; i < LT * DDIM; i += 128) {
        int li = i >> 7;             // /128: layer
        int d  = i & (DDIM - 1);
        GI* src = (GI*)(xrow + (size_t)li * ((size_t)NPOS * DDIM) + d);
        LI* dst = (LI*)&Xl[(li >> 1) * PSTRIDE + (d + 1) * 2 + (li & 1)];
        __builtin_amdgcn_global_load_async_to_lds_b32(src, dst, 0, 0);
    }
    if (tid < LT) {                  // zero halo columns (d = -1 and d = 128)
        Xl[(tid >> 1) * PSTRIDE + 0 * 2 + (tid & 1)]   = 0.0f;
        Xl[(tid >> 1) * PSTRIDE + 129 * 2 + (tid & 1)] = 0.0f;
    }
    wait_async_zero();
    __syncthreads();
#else
    for (int i = tid; i < LT * DDIM; i += 128) {
        int li = i >> 7;             // /128: layer
        int d  = i & (DDIM - 1);
        Xl[(li >> 1) * PSTRIDE + (d + 1) * 2 + (li & 1)] =
            xrow[(size_t)li * ((size_t)NPOS * DDIM) + d];
    }
    if (tid < LT) {                  // zero halo columns (d = -1 and d = 128)
        Xl[(tid >> 1) * PSTRIDE + 0 * 2 + (tid & 1)]   = 0.0f;
        Xl[(tid >> 1) * PSTRIDE + 129 * 2 + (tid & 1)] = 0.0f;
    }
    __syncthreads();
#endif

    const int wave   = tid >> 5;
    const int lane   = tid & 31;
    const int laneLo = lane & 15;
    const int hiHalf = lane >> 4;    // 0: lanes 0-15, 1: lanes 16-31

    const int mtile = wave >> 1;                 // 0 or 1 (C rows 0-15 / 16-31)
    const int row   = mtile * 16 + laneLo;       // A-matrix M index for this lane
    // Rows >= 28 only feed C rows that are never stored; just clamp for safety
    // (no mask multiply needed -> no VALU in the WMMA loop).
    const int rowc  = (row < LS) ? row : (LS - 1);

    // ---- preload 27 A fragments: (tap t, K-chunk kc) -> w[row][kc*4+..][t] ----
    // A 16x4 f32 layout: vgpr0 = K0 (lanes0-15) / K2 (lanes16-31); vgpr1 = K1 / K3
    v2f afrag[KSZ * 9];
#pragma unroll
    for (int t = 0; t < KSZ; ++t) {
#pragma unroll
        for (int kc = 0; kc < 9; ++kc) {
            const int k0 = kc * 4 + hiHalf * 2;
            const float* wp = w + ((size_t)rowc * LT + k0) * KSZ + t;
            v2f a;
            a.x = wp[0];      // w[row][k0  ][t]
            a.y = wp[KSZ];    // w[row][k0+1][t]
            afrag[t * 9 + kc] = a;
        }
    }

    // ---- 4 d-tiles per wave: C[16x16] = sum over 3 taps x 9 K-chunks ----
    const int dt0 = (wave & 1) * 4;
#pragma unroll
    for (int dt = 0; dt < 4; ++dt) {
        const int dtile = dt0 + dt;
        const int d = dtile * 16 + laneLo;       // output column for this lane
        v8f acc = {};
#pragma unroll
        for (int t = 0; t < KSZ; ++t) {
#pragma unroll
            for (int kc = 0; kc < 9; ++kc) {
                const int kpair = kc * 2 + hiHalf;   // layer pair {2kp, 2kp+1}
                // padded col index: (d + t - 1) + 1 = d + t  in [0, 129]
                // single aligned 8B LDS load -> {b.x, b.y} in adjacent VGPRs
                const v2f b = *(const v2f*)&Xl[kpair * PSTRIDE + (d + t) * 2];
                acc = __builtin_amdgcn_wmma_f32_16x16x4_f32(
                    false, afrag[t * 9 + kc], false, b,
                    (short)0, acc, false, false);
            }
        }
        // C layout: vgpr r -> M = mtile*16 + r (lanes0-15) / + r + 8 (lanes16-31)
#pragma unroll
        for (int r = 0; r < 8; ++r) {
            const int lo = mtile * 16 + r + hiHalf * 8;
            if (lo < LS) {
                o[(size_t)lo * ((size_t)NPOS * DDIM) + (size_t)n * DDIM + d] = acc[r];
            }
        }
    }
}

extern "C" void kernel_launch(void* const* d_in, const int* in_sizes, int n_in,
                              void* d_out, int out_size, void* d_ws, size_t ws_size,
                              hipStream_t stream) {
    const float* ks = (const float*)d_in[0];
    const float* vs = (const float*)d_in[1];
    const float* wk = (const float*)d_in[2];
    const float* wv = (const float*)d_in[3];
    float* out = (float*)d_out;

    dim3 grid(2 * NPOS);   // (tensor, n) pairs
    dim3 block(128);       // 4 waves
    hipLaunchKernelGGL(blend_wmma_f32, grid, block, 0, stream,
                       ks, vs, wk, wv, out);
}